// DecoderAttentionRNN_90340342104528
// MI455X (gfx1250) — compile-verified
//
#include <hip/hip_runtime.h>
#include <math.h>

#define HD   1000
#define VOC  50257
#define NL   4
#define G4H  4000
#define WIN  5        // 2D+1
#define MAXLEN 50

typedef float v2f __attribute__((ext_vector_type(2)));
typedef float v8f __attribute__((ext_vector_type(8)));

// One 16x16x4 f32 WMMA step. A-tile per-lane layout (ISA 7.12.2):
//   lane L supplies A[M = L&15][k0 + 2*(L>>4) + {0,1}]  -> one b64 load
// B is the GEMV vector broadcast into all 16 N columns:
//   lane L supplies x[k0 + 2*(L>>4) + {0,1}]            -> broadcast b64 load
// Result: every lane's acc[r] = dot(A row (tile*16 + r + 8*(L>>4)), x).
__device__ __forceinline__ v8f wmma_step(const float* __restrict__ arow,
                                         const float* __restrict__ bvec,
                                         v8f acc) {
    v2f a = *(const v2f*)arow;
    v2f b = *(const v2f*)bvec;
    return __builtin_amdgcn_wmma_f32_16x16x4_f32(false, a, false, b,
                                                 (short)0, acc, false, false);
}

__device__ __forceinline__ float sigf(float x) { return 1.0f / (1.0f + expf(-x)); }

// ---------------------------------------------------------------- x0 = [emb[token], ht_hat]
__global__ void build_x0_kernel(const int* __restrict__ token,
                                const float* __restrict__ emb,
                                const float* __restrict__ ht_hat,
                                float* __restrict__ x0) {
    int j = blockIdx.x * blockDim.x + threadIdx.x;
    if (j >= 2 * HD) return;
    if (j < HD) x0[j] = emb[(long)token[0] * HD + j];
    else        x0[j] = ht_hat[j - HD];
}

// ---------------------------------------------------------------- generic WMMA GEMV: y = act(A x + bias)
template <bool TANH>
__global__ void gemv_wmma_kernel(const float* __restrict__ A, int K,
                                 const float* __restrict__ x,
                                 const float* __restrict__ bias,
                                 float* __restrict__ y, int nrows) {
    int wave = (int)(blockIdx.x * blockDim.x + threadIdx.x) >> 5;
    int lane = threadIdx.x & 31;
    int ntiles = (nrows + 15) >> 4;
    if (wave >= ntiles) return;               // wave-uniform exit: EXEC stays all-1 for WMMA
    int m = wave * 16 + (lane & 15);
    if (m >= nrows) m = nrows - 1;            // clamped duplicate row; never stored
    int kh = (lane >> 4) << 1;
    const float* arow = A + (long)m * K + kh;
    const float* xv   = x + kh;
    v8f acc = {};
#pragma unroll 2
    for (int k = 0; k < K; k += 4)
        acc = wmma_step(arow + k, xv + k, acc);
    if ((lane & 15) == 0) {                   // lanes 0 & 16 hold the two row halves
        int base = wave * 16 + ((lane >> 4) << 3);
#pragma unroll
        for (int r = 0; r < 8; ++r) {
            int row = base + r;
            if (row < nrows) {
                float v = acc[r] + bias[row];
                y[row] = TANH ? tanhf(v) : v;
            }
        }
    }
}

// ---------------------------------------------------------------- LSTM gates: g = Wih*x + Whh*h + bih + bhh
__global__ void lstm_gates_kernel(const float* __restrict__ Wih, int Kx,
                                  const float* __restrict__ x,
                                  const float* __restrict__ Whh,
                                  const float* __restrict__ hprev,
                                  const float* __restrict__ bih,
                                  const float* __restrict__ bhh,
                                  float* __restrict__ gates) {
    int wave = (int)(blockIdx.x * blockDim.x + threadIdx.x) >> 5;
    int lane = threadIdx.x & 31;
    if (wave >= (G4H / 16)) return;
    int m  = wave * 16 + (lane & 15);         // 4000 % 16 == 0, always in range
    int kh = (lane >> 4) << 1;
    v8f acc = {};
    const float* a0 = Wih + (long)m * Kx + kh;
    const float* x0 = x + kh;
#pragma unroll 2
    for (int k = 0; k < Kx; k += 4)
        acc = wmma_step(a0 + k, x0 + k, acc);
    const float* a1 = Whh + (long)m * HD + kh;
    const float* h0 = hprev + kh;
#pragma unroll 2
    for (int k = 0; k < HD; k += 4)
        acc = wmma_step(a1 + k, h0 + k, acc);
    if ((lane & 15) == 0) {
        int base = wave * 16 + ((lane >> 4) << 3);
#pragma unroll
        for (int r = 0; r < 8; ++r) {
            int row = base + r;
            gates[row] = acc[r] + bih[row] + bhh[row];
        }
    }
}

// ---------------------------------------------------------------- LSTM cell (elementwise)
__global__ void lstm_cell_kernel(const float* __restrict__ gates,
                                 const float* __restrict__ Cin,
                                 float* __restrict__ h_ws,
                                 float* __restrict__ h_out,
                                 float* __restrict__ c_out) {
    int j = blockIdx.x * blockDim.x + threadIdx.x;
    if (j >= HD) return;
    float i = gates[j];
    float f = gates[HD + j];
    float g = gates[2 * HD + j];
    float o = gates[3 * HD + j];
    float c = sigf(f) * Cin[j] + sigf(i) * tanhf(g);
    float h = sigf(o) * tanhf(c);
    h_ws[j] = h; h_out[j] = h; c_out[j] = c;
}

// ---------------------------------------------------------------- Pt / lb + zero scores
__global__ void pt_kernel(const float* __restrict__ t,
                          const float* __restrict__ Vp_w,
                          const float* __restrict__ Vp_b,
                          float* __restrict__ pt_f,
                          int* __restrict__ lb_i,
                          float* __restrict__ scores) {
    __shared__ float sm[256];
    int tid = threadIdx.x;
    float s = 0.0f;
    for (int i = tid; i < HD; i += 256) s += t[i] * Vp_w[i];
    sm[tid] = s; __syncthreads();
    for (int w = 128; w > 0; w >>= 1) {
        if (tid < w) sm[tid] += sm[tid + w];
        __syncthreads();
    }
    if (tid == 0) {
        float a  = sm[0] + Vp_b[0];
        float Pt = floorf((float)MAXLEN * sigf(a));
        Pt = fminf(fmaxf(Pt, 2.0f), (float)(MAXLEN - 3));
        *pt_f = Pt;
        *lb_i = (int)(Pt - 2.0f);
    }
    if (tid < WIN) scores[tid] = 0.0f;
}

// ---------------------------------------------------------------- attention scores via WMMA
// D[j-tile, s] = Wal_w(16xK) * enc_window^T(Kx5); fold h_top, atomicAdd to scores.
__global__ void attn_scores_kernel(const float* __restrict__ Wal_w,
                                   const float* __restrict__ Wal_b,
                                   const float* __restrict__ enc,
                                   const float* __restrict__ h_top,
                                   const int* __restrict__ lb_i,
                                   float* __restrict__ scores) {
    int wave = (int)(blockIdx.x * blockDim.x + threadIdx.x) >> 5;
    int lane = threadIdx.x & 31;
    const int ntiles = (HD + 15) >> 4;        // 63
    if (wave >= ntiles) return;
    int lb = *lb_i;
    int N  = lane & 15;
    int kh = (lane >> 4) << 1;
    int m  = wave * 16 + N;
    if (m >= HD) m = HD - 1;                  // clamp; tail rows masked at accumulation
    int e = lb + (N < WIN ? N : WIN - 1);     // clamp junk columns to a valid enc row
    const float* arow = Wal_w + (long)m * HD + kh;
    const float* brow = enc + (long)e * HD + kh;
    v8f acc = {};
#pragma unroll 2
    for (int k = 0; k < HD; k += 4)
        acc = wmma_step(arow + k, brow + k, acc);
    if (N < WIN) {
        int base = wave * 16 + ((lane >> 4) << 3);
        float p = 0.0f;
#pragma unroll
        for (int r = 0; r < 8; ++r) {
            int j = base + r;
            if (j < HD) p += h_top[j] * (acc[r] + Wal_b[j]);
        }
        atomicAdd(&scores[N], p);
    }
}

// ---------------------------------------------------------------- softmax * gaussian decay
__global__ void align_kernel(const float* __restrict__ scores,
                             const float* __restrict__ pt_f,
                             const int* __restrict__ lb_i,
                             float* __restrict__ align) {
    if (threadIdx.x != 0) return;
    float Pt = *pt_f;
    int lb = *lb_i;
    float mx = scores[0];
    for (int s = 1; s < WIN; ++s) mx = fmaxf(mx, scores[s]);
    float e[WIN], sum = 0.0f;
    for (int s = 0; s < WIN; ++s) { e[s] = expf(scores[s] - mx); sum += e[s]; }
    for (int s = 0; s < WIN; ++s) {
        float d = (float)(lb + s) - Pt;
        align[s] = (e[s] / sum) * expf(-(d * d) * 0.5f);  // sigma = D/2 = 1
    }
}

// ---------------------------------------------------------------- ht_new (elementwise)
__global__ void ht_kernel(const float* __restrict__ align,
                          const int* __restrict__ lb_i,
                          const float* __restrict__ enc,
                          const float* __restrict__ h_top,
                          const float* __restrict__ Wc_w,
                          const float* __restrict__ Wc_b,
                          float* __restrict__ ht_ws,
                          float* __restrict__ ht_out) {
    int j = blockIdx.x * blockDim.x + threadIdx.x;
    if (j >= HD) return;
    int lb = *lb_i;
    float acc = Wc_w[WIN] * h_top[j] + Wc_b[0];
#pragma unroll
    for (int s = 0; s < WIN; ++s)
        acc += Wc_w[s] * align[s] * enc[(long)(lb + s) * HD + j];
    float v = tanhf(acc);
    ht_ws[j] = v; ht_out[j] = v;
}

// ---------------------------------------------------------------- log_softmax stats (single block)
__global__ void lse_kernel(const float* __restrict__ logits, float* __restrict__ stats) {
    __shared__ float sm[1024];
    int tid = threadIdx.x;
    float m = -INFINITY;
    for (int i = tid; i < VOC; i += 1024) m = fmaxf(m, logits[i]);
    sm[tid] = m; __syncthreads();
    for (int w = 512; w > 0; w >>= 1) {
        if (tid < w) sm[tid] = fmaxf(sm[tid], sm[tid + w]);
        __syncthreads();
    }
    float mx = sm[0]; __syncthreads();
    float s = 0.0f;
    for (int i = tid; i < VOC; i += 1024) s += expf(logits[i] - mx);
    sm[tid] = s; __syncthreads();
    for (int w = 512; w > 0; w >>= 1) {
        if (tid < w) sm[tid] += sm[tid + w];
        __syncthreads();
    }
    if (tid == 0) { stats[0] = mx; stats[1] = logf(sm[0]); }
}

__global__ void logsoftmax_kernel(const float* __restrict__ logits,
                                  const float* __restrict__ stats,
                                  float* __restrict__ out) {
    int i = blockIdx.x * blockDim.x + threadIdx.x;
    if (i >= VOC) return;
    out[i] = logits[i] - stats[0] - stats[1];
}

// ---------------------------------------------------------------- host side
extern "C" void kernel_launch(void* const* d_in, const int* in_sizes, int n_in,
                              void* d_out, int out_size, void* d_ws, size_t ws_size,
                              hipStream_t stream) {
    const int*   token  = (const int*)d_in[0];
    const float* hidden = (const float*)d_in[1];
    const float* Cst    = (const float*)d_in[2];
    const float* ht_hat = (const float*)d_in[3];
    const float* enc    = (const float*)d_in[4];
    const float* emb    = (const float*)d_in[5];
    const float* Wih0   = (const float*)d_in[6];
    const float* WihR   = (const float*)d_in[7];
    const float* Whh    = (const float*)d_in[8];
    const float* bih    = (const float*)d_in[9];
    const float* bhh    = (const float*)d_in[10];
    const float* Wp_w   = (const float*)d_in[11];
    const float* Wp_b   = (const float*)d_in[12];
    const float* Vp_w   = (const float*)d_in[13];
    const float* Vp_b   = (const float*)d_in[14];
    const float* Wal_w  = (const float*)d_in[15];
    const float* Wal_b  = (const float*)d_in[16];
    const float* Wc_w   = (const float*)d_in[17];
    const float* Wc_b   = (const float*)d_in[18];
    const float* Ws_w   = (const float*)d_in[19];
    const float* Ws_b   = (const float*)d_in[20];

    float* out = (float*)d_out;
    float* out_logsm = out;                    // [VOC]
    float* out_hid   = out + VOC;              // [NL*HD]
    float* out_C     = out + VOC + NL * HD;    // [NL*HD]
    float* out_ht    = out + VOC + 2 * NL * HD;// [HD]

    // workspace layout (floats)
    float* ws      = (float*)d_ws;
    float* x0_ws   = ws;                       // 2000
    float* gates   = ws + 2048;                // 4000
    float* h_ws    = ws + 6144;                // 4 x 1024
    float* t_ws    = ws + 10240;               // 1000
    float* pt_f    = ws + 11264;               // 1
    int*   lb_i    = (int*)(ws + 11265);       // 1
    float* scores  = ws + 11268;               // 5
    float* align   = ws + 11280;               // 5
    float* stats   = ws + 11296;               // 2
    float* logits  = ws + 11392;               // 50257
    if (ws_size < (size_t)(11392 + VOC) * sizeof(float)) return;

    // 1) input-feeding concat
    build_x0_kernel<<<8, 256, 0, stream>>>(token, emb, ht_hat, x0_ws);

    // 2) 4 LSTM layers (sequential dependency)
    for (int l = 0; l < NL; ++l) {
        const float* Wih = (l == 0) ? Wih0 : (WihR + (long)(l - 1) * G4H * HD);
        int Kx           = (l == 0) ? 2 * HD : HD;
        const float* xin = (l == 0) ? x0_ws : (h_ws + (l - 1) * 1024);
        lstm_gates_kernel<<<32, 256, 0, stream>>>(
            Wih, Kx, xin, Whh + (long)l * G4H * HD, hidden + l * HD,
            bih + l * G4H, bhh + l * G4H, gates);
        lstm_cell_kernel<<<4, 256, 0, stream>>>(
            gates, Cst + l * HD, h_ws + l * 1024,
            out_hid + l * HD, out_C + l * HD);
    }
    const float* h_top = h_ws + (NL - 1) * 1024;

    // 3) predictive position Pt
    gemv_wmma_kernel<true><<<8, 256, 0, stream>>>(Wp_w, HD, h_top, Wp_b, t_ws, HD);
    pt_kernel<<<1, 256, 0, stream>>>(t_ws, Vp_w, Vp_b, pt_f, lb_i, scores);

    // 4) local attention scores + align
    attn_scores_kernel<<<8, 256, 0, stream>>>(Wal_w, Wal_b, enc, h_top, lb_i, scores);
    align_kernel<<<1, 32, 0, stream>>>(scores, pt_f, lb_i, align);
    ht_kernel<<<4, 256, 0, stream>>>(align, lb_i, enc, h_top, Wc_w, Wc_b, t_ws, out_ht);

    // 5) vocab projection (the 201 MB stream) + log_softmax
    gemv_wmma_kernel<false><<<393, 256, 0, stream>>>(Ws_w, HD, t_ws, Ws_b, logits, VOC);
    lse_kernel<<<1, 1024, 0, stream>>>(logits, stats);
    logsoftmax_kernel<<<(VOC + 255) / 256, 256, 0, stream>>>(logits, stats, out_logsm);
}